// PPN_71768903516625
// MI455X (gfx1250) — compile-verified
//
#include <hip/hip_runtime.h>

// ---------------------------------------------------------------------------
// Sparse voxel mask-align / prune / merge / 2x-downsample for MI455X (gfx1250).
//
// Pure HBM-streaming workload (~1.1 GB moved, ~0.2 GFLOP) => target is the
// 23.3 TB/s HBM path, not the matrix units. Design:
//   * packed open-addressing hash tables (key<<IDX_BITS | idx) in d_ws,
//     min-index duplicate resolution via 64-bit atomicMin (matches JAX's
//     stable argsort + searchsorted semantics exactly)
//   * wave32 parallel probe: 32 slots per coalesced load, v_ballot_w32 to
//     find first match/empty, ds_bpermute_b32 to broadcast the hit index
//   * tables (36 MB) stay resident in the 192 MB L2; streamed outputs use
//     nontemporal stores so they don't evict the tables
//   * one wave per 64-wide feature row, float2 per lane (256 B coalesced)
// ---------------------------------------------------------------------------

typedef __attribute__((ext_vector_type(2))) float v2f;
typedef __attribute__((ext_vector_type(4))) int   v4i;

#define EMPTY64 0xFFFFFFFFFFFFFFFFull

__device__ __forceinline__ unsigned long long mix64(unsigned long long x) {
  // splitmix64 finalizer
  x ^= x >> 30; x *= 0xbf58476d1ce4e5b9ull;
  x ^= x >> 27; x *= 0x94d049bb133111ebull;
  x ^= x >> 31;
  return x;
}

// key = ((b*1024 + x)*1024 + y)*1024 + z  ; coords < 1024 => key < 2^40
__device__ __forceinline__ unsigned long long key4(v4i c) {
  return ((((unsigned long long)(unsigned)c.x * 1024ull
          + (unsigned)c.y) * 1024ull
          + (unsigned)c.z) * 1024ull
          + (unsigned)c.w);
}

// ---------------------------------------------------------------------------
// Build: insert (key -> min original index) into packed open-addressing table.
// Entry = (key << IDX_BITS) | index  (< 2^61, so EMPTY64 is unambiguous).
// ---------------------------------------------------------------------------
template <int IDX_BITS>
__global__ void __launch_bounds__(256)
k_build(const int* __restrict__ coords, int n,
        unsigned long long* __restrict__ tab, unsigned tmask) {
  int i = blockIdx.x * blockDim.x + threadIdx.x;
  if (i >= n) return;
  v4i c = ((const v4i*)coords)[i];
  unsigned long long key   = key4(c);
  unsigned long long entry = (key << IDX_BITS) | (unsigned long long)(unsigned)i;
  unsigned slot = (unsigned)mix64(key) & tmask;
  for (;;) {
    unsigned long long prev = atomicCAS(&tab[slot], EMPTY64, entry);
    if (prev == EMPTY64) return;                 // claimed empty slot
    if ((prev >> IDX_BITS) == key) {             // same key: keep min index
      atomicMin(&tab[slot], entry);              // same high bits => min(idx)
      return;
    }
    slot = (slot + 1) & tmask;                   // different key: keep probing
  }
}

// ---------------------------------------------------------------------------
// Wave32 cooperative probe: lane L inspects slot (h+L). One coalesced 256 B
// load covers the whole probe window; ballot finds the first match-or-empty
// slot in linear-probe order. Returns index (>=0) on hit, -1 on miss.
// ---------------------------------------------------------------------------
template <int IDX_BITS>
__device__ __forceinline__ int wave_probe(const unsigned long long* __restrict__ tab,
                                          unsigned tmask, unsigned long long key,
                                          int lane) {
  unsigned h = (unsigned)mix64(key);
  for (int it = 0; it < 64; ++it) {              // bound; tables are < 75% full
    unsigned long long e = tab[(h + (unsigned)lane) & tmask];
    int match = (e != EMPTY64) && ((e >> IDX_BITS) == key);
    unsigned mm   = __builtin_amdgcn_ballot_w32(match);
    unsigned em   = __builtin_amdgcn_ballot_w32(e == EMPTY64);
    unsigned stop = mm | em;
    if (stop) {                                  // wave-uniform branch
      unsigned first = (unsigned)__builtin_ctz(stop);
      int idx = (int)(unsigned)(e & ((1ull << IDX_BITS) - 1ull));
      int got = __builtin_amdgcn_ds_bpermute((int)(first << 2), idx);
      return ((mm >> first) & 1u) ? got : -1;    // empty before match => miss
    }
    h += 32;
  }
  return -1;
}

// ---------------------------------------------------------------------------
// Main fused kernel: one wave per x row (64 f32 channels, float2 per lane).
// ---------------------------------------------------------------------------
__global__ void __launch_bounds__(256)
k_main(const int* __restrict__ x_coords, const float* __restrict__ x_feats,
       const float* __restrict__ mask_scores,
       const unsigned long long* __restrict__ mtab, unsigned mt_mask,
       const unsigned long long* __restrict__ dtab, unsigned dt_mask,
       float* __restrict__ o_pruned, float* __restrict__ o_scaled,
       float* __restrict__ o_merged, float* __restrict__ o_dsmask, int N) {
  const int lane = threadIdx.x & 31;
  const int row  = blockIdx.x * 8 + (threadIdx.x >> 5);
  if (row >= N) return;                          // wave-uniform exit

  // issue streaming loads early; probe latency overlaps them
  v4i c = __builtin_nontemporal_load((const v4i*)x_coords + row);
  v2f f = __builtin_nontemporal_load((const v2f*)(x_feats + (size_t)row * 64) + lane);

  unsigned long long key = key4(c);
  int mi = wave_probe<21>(mtab, mt_mask, key, lane);

  float s0 = 0.f, s1 = 0.f;                      // m_at_x (0 on miss)
  if (mi >= 0) {
    v2f ms = *((const v2f*)mask_scores + mi);
    s0 = ms.x; s1 = ms.y;
  }
  float attn = (s1 > 0.5f) ? 1.f : 0.f;          // THRESH = 0.5, strict >
  float sc2  = s1 * attn;                        // score_at_x * attn

  v2f pr; pr.x = f.x * attn; pr.y = f.y * attn;
  v2f sv; sv.x = f.x * sc2;  sv.y = f.y * sc2;
  __builtin_nontemporal_store(pr, (v2f*)(o_pruned + (size_t)row * 64) + lane);
  __builtin_nontemporal_store(sv, (v2f*)(o_scaled + (size_t)row * 64) + lane);

  float* mrow = o_merged + (size_t)row * 66;     // 264 B rows; 8 B aligned
  if (lane == 0) {
    v2f m2; m2.x = s0; m2.y = s1;
    __builtin_nontemporal_store(m2, (v2f*)mrow);
  }
  __builtin_nontemporal_store(f, (v2f*)(mrow + 2) + lane);

  if (attn != 0.f) {                             // wave-uniform
    v4i p; p.x = c.x; p.y = c.y >> 1; p.z = c.z >> 1; p.w = c.w >> 1;
    int di = wave_probe<18>(dtab, dt_mask, key4(p), lane);
    if (di >= 0 && lane == 0) o_dsmask[di] = 1.0f;  // segment_max of {0,1}
  }
}

// ---------------------------------------------------------------------------
extern "C" void kernel_launch(void* const* d_in, const int* in_sizes, int n_in,
                              void* d_out, int out_size, void* d_ws, size_t ws_size,
                              hipStream_t stream) {
  const int*   x_coords    = (const int*)d_in[0];
  const float* x_feats     = (const float*)d_in[1];
  const int*   mask_coords = (const int*)d_in[2];
  const float* mask_scores = (const float*)d_in[3];
  const int*   ds_coords   = (const int*)d_in[4];

  const int N  = in_sizes[1] / 64;   // 1,000,000
  const int M  = in_sizes[3] / 2;    // 1,500,000
  const int Nd = in_sizes[4] / 4;    //   250,000

  // d_ws layout: [4M-slot mask table | 512K-slot ds table] = 36 MB (L2-resident)
  const unsigned MT_SLOTS = 1u << 22;   // load factor 0.36, idx fits 21 bits
  const unsigned DT_SLOTS = 1u << 19;   // load factor 0.48, idx fits 18 bits
  unsigned long long* mtab = (unsigned long long*)d_ws;
  unsigned long long* dtab = mtab + MT_SLOTS;

  float* out      = (float*)d_out;
  float* o_pruned = out;
  float* o_scaled = out + (size_t)N * 64;
  float* o_merged = out + (size_t)N * 128;
  float* o_dsmask = out + (size_t)N * 128 + (size_t)N * 66;

  // empty sentinel = all-ones bytes; ds_mask must start at 0 each call
  hipMemsetAsync(d_ws, 0xFF, (size_t)(MT_SLOTS + DT_SLOTS) * 8, stream);
  hipMemsetAsync(o_dsmask, 0, (size_t)Nd * sizeof(float), stream);

  k_build<21><<<(M  + 255) / 256, 256, 0, stream>>>(mask_coords, M,  mtab, MT_SLOTS - 1);
  k_build<18><<<(Nd + 255) / 256, 256, 0, stream>>>(ds_coords,   Nd, dtab, DT_SLOTS - 1);

  k_main<<<(N + 7) / 8, 256, 0, stream>>>(x_coords, x_feats, mask_scores,
                                          mtab, MT_SLOTS - 1, dtab, DT_SLOTS - 1,
                                          o_pruned, o_scaled, o_merged, o_dsmask, N);
}